// PureQNet_87960930222355
// MI455X (gfx1250) — compile-verified
//
#include <hip/hip_runtime.h>

typedef float v2f __attribute__((ext_vector_type(2)));
typedef float v8f __attribute__((ext_vector_type(8)));

#define SAMP_PER_BLOCK 16

__global__ __launch_bounds__(256) void qnet_fused_kernel(
    const float* __restrict__ x,        // (B,1,64,64)
    const float* __restrict__ qw,       // (6,4)
    const float* __restrict__ hw,       // (10,10) row-major head_w
    const float* __restrict__ hb,       // (10,)
    const float* __restrict__ scale_p,  // (1,)
    float* __restrict__ out)            // (B,10)
{
    __shared__ float pooledLDS[16 * 16];  // [local sample][16 pooled sums]
    __shared__ float probsLDS[16 * 16];   // [sample row][16 probs]

    const int tid  = threadIdx.x;
    const int lane = tid & 31;
    const int wv   = tid >> 5;                       // 8 waves
    const int base = blockIdx.x * SAMP_PER_BLOCK;    // first sample of block

    // ---------------- Phase 1: pooling (sum over 16x16 blocks) ----------------
    // wave wv pools local samples 2wv and 2wv+1; lane l loads float4 #(l+32k).
    // For that pattern: colblock = (l&15)>>2 (fixed per lane), rowblock = k>>3.
    #pragma unroll
    for (int ss = 0; ss < 2; ++ss) {
        const int sl = wv * 2 + ss;  // local sample 0..15
        const float4* __restrict__ xs =
            (const float4*)(x + (size_t)(base + sl) * 4096);
        float acc[4];
        #pragma unroll
        for (int rb = 0; rb < 4; ++rb) {
            float s = 0.0f;
            #pragma unroll
            for (int k8 = 0; k8 < 8; ++k8) {
                const int k = rb * 8 + k8;
                const float4 v = xs[lane + 32 * k];   // global_load_b128, coalesced
                s += v.x + v.y + v.z + v.w;
            }
            acc[rb] = s;
        }
        // reduce across the 8 lanes sharing a colblock: lanes {4cb..4cb+3} U {16+...}
        #pragma unroll
        for (int rb = 0; rb < 4; ++rb) {
            float v = acc[rb];
            v += __shfl_xor(v, 1);
            v += __shfl_xor(v, 2);
            v += __shfl_xor(v, 16);
            acc[rb] = v;
        }
        if (((lane & 3) == 0) && lane < 16) {
            const int cb = lane >> 2;
            #pragma unroll
            for (int rb = 0; rb < 4; ++rb)
                pooledLDS[sl * 16 + rb * 4 + cb] = acc[rb];  // /256 cancels in normalize
        }
    }
    __syncthreads();

    // ---------------- Phase 2: circuit + WMMA head (wave 0 only) ----------------
    if (tid < 32) {
        const float scale = scale_p[0];
        const int sm = lane & 15;   // sample row (0..15); lanes 16-31 redo row sm
        const int hi = lane >> 4;   // 0: K lo pair, 1: K hi pair (A/B fragment halves)

        // load pooled row, normalize (AmplitudeEmbedding)
        float a[16];
        float nrm = 0.0f;
        #pragma unroll
        for (int i = 0; i < 16; ++i) { a[i] = pooledLDS[sm * 16 + i]; nrm += a[i] * a[i]; }
        const float inv = 1.0f / sqrtf(nrm);

        float re[16], im[16];
        #pragma unroll
        for (int i = 0; i < 16; ++i) { re[i] = a[i] * inv; im[i] = 0.0f; }

        // 6 layers: RX(q0..q3) then CNOT ring. qubit q <-> index bit (3-q).
        #pragma unroll
        for (int l = 0; l < 6; ++l) {
            #pragma unroll
            for (int q = 0; q < 4; ++q) {
                const float half = 0.5f * qw[l * 4 + q];
                const float c = cosf(half);
                const float s = sinf(half);
                const int m = 1 << (3 - q);
                #pragma unroll
                for (int i = 0; i < 16; ++i) {
                    if (!(i & m)) {
                        const int j = i | m;
                        const float r0 = re[i], i0 = im[i];
                        const float r1 = re[j], i1 = im[j];
                        re[i] = c * r0 + s * i1;   // psi' = c*psi - i*s*psi_pair
                        im[i] = c * i0 - s * r1;
                        re[j] = c * r1 + s * i0;
                        im[j] = c * i1 - s * r0;
                    }
                }
            }
            #pragma unroll
            for (int q = 0; q < 4; ++q) {          // CNOT(q, (q+1)%4)
                const int cm = 1 << (3 - q);
                const int tm = 1 << (3 - ((q + 1) & 3));
                #pragma unroll
                for (int i = 0; i < 16; ++i) {
                    if ((i & cm) && !(i & tm)) {
                        const int j = i | tm;
                        float t;
                        t = re[i]; re[i] = re[j]; re[j] = t;
                        t = im[i]; im[i] = im[j]; im[j] = t;
                    }
                }
            }
        }

        // probs -> LDS (row sm); lanes 16-31 hold duplicates, don't write
        if (lane < 16) {
            #pragma unroll
            for (int i = 0; i < 16; ++i)
                probsLDS[lane * 16 + i] = re[i] * re[i] + im[i] * im[i];
        }
        __asm__ volatile("s_wait_dscnt 0" ::: "memory");  // same-wave DS ordering

        // A fragments: 16x4 f32 layout — VGPR0: K=4t+2hi, VGPR1: K=4t+2hi+1, row=sm
        v2f Af[4];
        #pragma unroll
        for (int t = 0; t < 4; ++t) {
            Af[t].x = probsLDS[sm * 16 + 4 * t + 2 * hi];
            Af[t].y = probsLDS[sm * 16 + 4 * t + 2 * hi + 1];
        }

        // B fragments: M[k][n] = sum_f sign(k,f) * head_w[n][f]; n = sm, zero for n>=10
        const int n = sm;
        const bool valid = (n < 10);
        float wn[10];
        #pragma unroll
        for (int j = 0; j < 10; ++j) wn[j] = valid ? hw[n * 10 + j] : 0.0f;
        const int cmask[10] = {8, 4, 2, 1, 12, 10, 9, 6, 5, 3};  // index-bit masks
        v2f Bf[4];
        #pragma unroll
        for (int t = 0; t < 4; ++t) {
            const int k0 = 4 * t + 2 * hi;
            float m0 = 0.0f, m1 = 0.0f;
            #pragma unroll
            for (int j = 0; j < 10; ++j) {
                const float w = wn[j];
                m0 += (__popc(k0 & cmask[j]) & 1) ? -w : w;
                m1 += (__popc((k0 + 1) & cmask[j]) & 1) ? -w : w;
            }
            Bf[t].x = m0;
            Bf[t].y = m1;
        }

        // D(16x16) = probs(16x16) @ M(16x16), K chained in 4 slices of 4
        v8f acc = {};
        acc = __builtin_amdgcn_wmma_f32_16x16x4_f32(false, Af[0], false, Bf[0],
                                                    (short)0, acc, false, false);
        acc = __builtin_amdgcn_wmma_f32_16x16x4_f32(false, Af[1], false, Bf[1],
                                                    (short)0, acc, false, false);
        acc = __builtin_amdgcn_wmma_f32_16x16x4_f32(false, Af[2], false, Bf[2],
                                                    (short)0, acc, false, false);
        acc = __builtin_amdgcn_wmma_f32_16x16x4_f32(false, Af[3], false, Bf[3],
                                                    (short)0, acc, false, false);

        // C/D layout: VGPR v -> sample row v + 8*hi, column n = lane&15
        if (valid) {
            const float b = hb[n];
            #pragma unroll
            for (int v = 0; v < 8; ++v) {
                const int samp = base + v + 8 * hi;
                out[samp * 10 + n] = (acc[v] + b) * scale;
            }
        }
    }
}

extern "C" void kernel_launch(void* const* d_in, const int* in_sizes, int n_in,
                              void* d_out, int out_size, void* d_ws, size_t ws_size,
                              hipStream_t stream) {
    const float* x  = (const float*)d_in[0];
    const float* qw = (const float*)d_in[1];
    const float* hw = (const float*)d_in[2];
    const float* hb = (const float*)d_in[3];
    const float* sc = (const float*)d_in[4];
    float* out = (float*)d_out;

    const int nsamp  = in_sizes[0] / 4096;        // 8192
    const int blocks = nsamp / SAMP_PER_BLOCK;    // 512

    qnet_fused_kernel<<<blocks, 256, 0, stream>>>(x, qw, hw, hb, sc, out);
}